// MoLEModule_60181081751966
// MI455X (gfx1250) — compile-verified
//
#include <hip/hip_runtime.h>
#include <hip/hip_bf16.h>

typedef __attribute__((ext_vector_type(2))) float v2f;
typedef __attribute__((ext_vector_type(8))) float v8f;

#define Bc    4
#define Sc    1024
#define INc   1024
#define OUTc  1024
#define Kc    8
#define Rc    16
#define BSc   (Bc * Sc)        /* 4096 rows */
#define KOc   (Kc * OUTc)      /* 8192 features per row */
#define FLATc ((size_t)Sc * KOc) /* 8388608 per-batch flat length */
#define SCALING 2.0f
#define LN_EPS  1e-6f

// D = A(16x4 f32) x B(4x16 f32) + C(16x16 f32), full wave32, native fp32 WMMA.
static __device__ __forceinline__ v8f wmma_f32(v2f a, v2f b, v8f c) {
  return __builtin_amdgcn_wmma_f32_16x16x4_f32(false, a, false, b, (short)0, c,
                                               false, false);
}

// ---------------------------------------------------------------------------
// Kernel 1: xa[k, m, r] = sum_i x[m,i] * A[k,r,i]   (R = 16 -> one N tile)
// grid(BS/16, K), block(32): one wave per (M-tile, expert).
// ---------------------------------------------------------------------------
__global__ __launch_bounds__(32) void k_xa(const float* __restrict__ x,
                                           const float* __restrict__ A,
                                           float* __restrict__ xa) {
  const int tm = blockIdx.x;
  const int k  = blockIdx.y;
  const int lane = threadIdx.x;
  const int l15 = lane & 15;
  const int lh  = (lane >> 4) << 1;          // K-pair base: 0 or 2
  const float* xrow = x + (size_t)(tm * 16 + l15) * INc;           // A: row M=l15
  const float* arow = A + ((size_t)k * Rc + l15) * INc;            // B: col N=l15
  v8f c = {};
  for (int kk = 0; kk < INc; kk += 4) {
    v2f a = *(const v2f*)(xrow + kk + lh);
    v2f b = *(const v2f*)(arow + kk + lh);
    c = wmma_f32(a, b, c);
  }
  float* out = xa + ((size_t)k * BSc + (size_t)tm * 16) * Rc;
  #pragma unroll
  for (int j = 0; j < 8; ++j) {
    const int m = j + ((lane >> 4) << 3);
    out[m * Rc + l15] = c[j];
  }
}

// ---------------------------------------------------------------------------
// Kernel 2: base[m, o] = sum_i x[m,i] * W[o,i]
// grid(BS/16, OUT/64), block(32): wave computes 16x64 (4 WMMA tiles, A reuse).
// ---------------------------------------------------------------------------
__global__ __launch_bounds__(32) void k_base(const float* __restrict__ x,
                                             const float* __restrict__ W,
                                             float* __restrict__ base) {
  const int tm = blockIdx.x;
  const int tn = blockIdx.y;                 // group of 64 output cols
  const int lane = threadIdx.x;
  const int l15 = lane & 15;
  const int lh  = (lane >> 4) << 1;
  const float* xrow = x + (size_t)(tm * 16 + l15) * INc;
  const float* w0 = W + (size_t)(tn * 64 +  0 + l15) * INc;
  const float* w1 = W + (size_t)(tn * 64 + 16 + l15) * INc;
  const float* w2 = W + (size_t)(tn * 64 + 32 + l15) * INc;
  const float* w3 = W + (size_t)(tn * 64 + 48 + l15) * INc;
  v8f c0 = {}, c1 = {}, c2 = {}, c3 = {};
  for (int kk = 0; kk < INc; kk += 4) {
    v2f a = *(const v2f*)(xrow + kk + lh);
    c0 = wmma_f32(a, *(const v2f*)(w0 + kk + lh), c0);
    c1 = wmma_f32(a, *(const v2f*)(w1 + kk + lh), c1);
    c2 = wmma_f32(a, *(const v2f*)(w2 + kk + lh), c2);
    c3 = wmma_f32(a, *(const v2f*)(w3 + kk + lh), c3);
  }
  #pragma unroll
  for (int j = 0; j < 8; ++j) {
    const int m = tm * 16 + j + ((lane >> 4) << 3);
    float* row = base + (size_t)m * OUTc + tn * 64 + l15;
    row[0] = c0[j]; row[16] = c1[j]; row[32] = c2[j]; row[48] = c3[j];
  }
}

// ---------------------------------------------------------------------------
// Kernel 3: E[k, m, o] = base[m,o] + SCALING * sum_r xa[k,m,r] * Bw[k,o,r]
// C initialized with base tile; A fragment pre-scaled by SCALING.
// grid(BS/16, OUT/64, K), block(32).
// ---------------------------------------------------------------------------
__global__ __launch_bounds__(32) void k_expert(const float* __restrict__ xa,
                                               const float* __restrict__ Bw,
                                               const float* __restrict__ base,
                                               float* __restrict__ E) {
  const int tm = blockIdx.x;
  const int tn = blockIdx.y;
  const int k  = blockIdx.z;
  const int lane = threadIdx.x;
  const int l15 = lane & 15;
  const int lh  = (lane >> 4) << 1;
  const float* xrow = xa + ((size_t)k * BSc + tm * 16 + l15) * Rc;
  const float* b0 = Bw + ((size_t)k * OUTc + tn * 64 +  0 + l15) * Rc;
  const float* b1 = Bw + ((size_t)k * OUTc + tn * 64 + 16 + l15) * Rc;
  const float* b2 = Bw + ((size_t)k * OUTc + tn * 64 + 32 + l15) * Rc;
  const float* b3 = Bw + ((size_t)k * OUTc + tn * 64 + 48 + l15) * Rc;
  v8f c0, c1, c2, c3;
  #pragma unroll
  for (int j = 0; j < 8; ++j) {
    const int m = tm * 16 + j + ((lane >> 4) << 3);
    const float* row = base + (size_t)m * OUTc + tn * 64 + l15;
    c0[j] = row[0]; c1[j] = row[16]; c2[j] = row[32]; c3[j] = row[48];
  }
  #pragma unroll
  for (int kk = 0; kk < Rc; kk += 4) {
    v2f a = *(const v2f*)(xrow + kk + lh);
    a = a * SCALING;                         // fold LoRA scaling into A fragment
    c0 = wmma_f32(a, *(const v2f*)(b0 + kk + lh), c0);
    c1 = wmma_f32(a, *(const v2f*)(b1 + kk + lh), c1);
    c2 = wmma_f32(a, *(const v2f*)(b2 + kk + lh), c2);
    c3 = wmma_f32(a, *(const v2f*)(b3 + kk + lh), c3);
  }
  float* eslab = E + (size_t)k * BSc * OUTc;
  #pragma unroll
  for (int j = 0; j < 8; ++j) {
    const int m = tm * 16 + j + ((lane >> 4) << 3);
    float* row = eslab + (size_t)m * OUTc + tn * 64 + l15;
    row[0] = c0[j]; row[16] = c1[j]; row[32] = c2[j]; row[48] = c3[j];
  }
}

// ---------------------------------------------------------------------------
// Kernel 4: LayerNorm stats per row m over the 8192-wide (k,o) feature dim.
// ---------------------------------------------------------------------------
__global__ __launch_bounds__(256) void k_lnstats(const float* __restrict__ E,
                                                 float* __restrict__ mu,
                                                 float* __restrict__ rstd) {
  __shared__ float ssum[256], ssq[256];
  const int m = blockIdx.x;
  const int t = threadIdx.x;
  float s = 0.f, q = 0.f;
  for (int idx = t; idx < KOc; idx += 256) {
    const int k = idx >> 10, o = idx & (OUTc - 1);
    const float e = E[((size_t)k * BSc + m) * OUTc + o];
    s += e; q += e * e;
  }
  ssum[t] = s; ssq[t] = q; __syncthreads();
  for (int off = 128; off > 0; off >>= 1) {
    if (t < off) { ssum[t] += ssum[t + off]; ssq[t] += ssq[t + off]; }
    __syncthreads();
  }
  if (t == 0) {
    const float mean = ssum[0] * (1.0f / KOc);
    const float var  = ssq[0] * (1.0f / KOc) - mean * mean;  // biased variance
    mu[m]   = mean;
    rstd[m] = rsqrtf(var + LN_EPS);
  }
}

// ---------------------------------------------------------------------------
// Kernel 5: per-row gate partials. partial[m, kp] = sum over 8192 features of
// LN(E)[m, j] * We[kp, s*KO + j]. We streamed non-temporally (268 MB, once per batch).
// ---------------------------------------------------------------------------
__global__ __launch_bounds__(256) void k_gate_partial(
    const float* __restrict__ E, const float* __restrict__ mu,
    const float* __restrict__ rstd, const float* __restrict__ gamma,
    const float* __restrict__ beta, const float* __restrict__ We,
    float* __restrict__ partial) {
  const int m = blockIdx.x;
  const int t = threadIdx.x;
  const int s = m & (Sc - 1);
  const float mean = mu[m], rs = rstd[m];
  float acc[Kc];
  #pragma unroll
  for (int kp = 0; kp < Kc; ++kp) acc[kp] = 0.f;
  for (int idx = t; idx < KOc; idx += 256) {
    const int k = idx >> 10, o = idx & (OUTc - 1);
    const float e = E[((size_t)k * BSc + m) * OUTc + o];
    const float v = (e - mean) * rs * gamma[idx] + beta[idx];
    const size_t f = (size_t)s * KOc + idx;
    __builtin_prefetch(We + f + 256, 0, 0);   // speculative stream-ahead
    #pragma unroll
    for (int kp = 0; kp < Kc; ++kp)
      acc[kp] += v * __builtin_nontemporal_load(We + (size_t)kp * FLATc + f);
  }
  __shared__ float red[256];
  for (int kp = 0; kp < Kc; ++kp) {
    red[t] = acc[kp]; __syncthreads();
    for (int off = 128; off > 0; off >>= 1) {
      if (t < off) red[t] += red[t + off];
      __syncthreads();
    }
    if (t == 0) partial[(size_t)m * Kc + kp] = red[0];
    __syncthreads();
  }
}

// ---------------------------------------------------------------------------
// Kernel 5b: deterministic reduce of partials over s -> logits[b, kp].
// ---------------------------------------------------------------------------
__global__ __launch_bounds__(256) void k_gate_reduce(const float* __restrict__ partial,
                                                     float* __restrict__ logits) {
  const int b = blockIdx.x, kp = blockIdx.y, t = threadIdx.x;
  float s = 0.f;
  for (int i = t; i < Sc; i += 256)
    s += partial[(size_t)(b * Sc + i) * Kc + kp];
  __shared__ float red[256];
  red[t] = s; __syncthreads();
  for (int off = 128; off > 0; off >>= 1) {
    if (t < off) red[t] += red[t + off];
    __syncthreads();
  }
  if (t == 0) logits[b * Kc + kp] = red[0];
}

// ---------------------------------------------------------------------------
// Kernel 6: softmax over K=8 experts per batch (tiny).
// ---------------------------------------------------------------------------
__global__ __launch_bounds__(32) void k_softmax(const float* __restrict__ logits,
                                                const float* __restrict__ tau,
                                                float* __restrict__ gates) {
  const int b = threadIdx.x;
  if (b < Bc) {
    const float it = 1.0f / fmaxf(tau[0], 1e-6f);
    float l[Kc], mx = -3.0e38f;
    #pragma unroll
    for (int k = 0; k < Kc; ++k) { l[k] = logits[b * Kc + k] * it; mx = fmaxf(mx, l[k]); }
    float sum = 0.f;
    #pragma unroll
    for (int k = 0; k < Kc; ++k) { l[k] = expf(l[k] - mx); sum += l[k]; }
    const float inv = 1.0f / sum;
    #pragma unroll
    for (int k = 0; k < Kc; ++k) gates[b * Kc + k] = l[k] * inv;
  }
}

// ---------------------------------------------------------------------------
// Kernel 7: out[m, o] = sum_k gates[b,k] * E[k, m, o]   (pre-norm E)
// ---------------------------------------------------------------------------
__global__ __launch_bounds__(256) void k_out(const float* __restrict__ E,
                                             const float* __restrict__ gates,
                                             float* __restrict__ out) {
  const size_t idx = (size_t)blockIdx.x * 256 + threadIdx.x;  // < BS*OUT
  const int m = (int)(idx >> 10);
  const int b = m >> 10;
  float acc = 0.f;
  #pragma unroll
  for (int k = 0; k < Kc; ++k)
    acc += gates[b * Kc + k] * E[(size_t)k * BSc * OUTc + idx];
  out[idx] = acc;
}

// ---------------------------------------------------------------------------
extern "C" void kernel_launch(void* const* d_in, const int* in_sizes, int n_in,
                              void* d_out, int out_size, void* d_ws, size_t ws_size,
                              hipStream_t stream) {
  (void)in_sizes; (void)n_in; (void)out_size; (void)ws_size;
  const float* x     = (const float*)d_in[0];  // [B,S,IN]
  const float* Wb    = (const float*)d_in[1];  // [OUT,IN]
  const float* A     = (const float*)d_in[2];  // [K,R,IN]
  const float* Bw    = (const float*)d_in[3];  // [K,OUT,R]
  const float* gamma = (const float*)d_in[4];  // [K*OUT]
  const float* beta  = (const float*)d_in[5];  // [K*OUT]
  const float* We    = (const float*)d_in[6];  // [K, S*K*OUT]
  const float* tau   = (const float*)d_in[7];  // scalar
  float* out = (float*)d_out;                  // [B,S,OUT]

  float* ws      = (float*)d_ws;
  float* xa      = ws;                               // 524288
  float* base    = xa + (size_t)Kc * BSc * Rc;       // 4194304
  float* E       = base + (size_t)BSc * OUTc;        // 33554432
  float* mu      = E + (size_t)Kc * BSc * OUTc;      // 4096
  float* rstd    = mu + BSc;                         // 4096
  float* partial = rstd + BSc;                       // 32768
  float* logits  = partial + (size_t)BSc * Kc;       // 32
  float* gates   = logits + Bc * Kc;                 // 32

  k_xa<<<dim3(BSc / 16, Kc), 32, 0, stream>>>(x, A, xa);
  k_base<<<dim3(BSc / 16, OUTc / 64), 32, 0, stream>>>(x, Wb, base);
  k_expert<<<dim3(BSc / 16, OUTc / 64, Kc), 32, 0, stream>>>(xa, Bw, base, E);
  k_lnstats<<<dim3(BSc), 256, 0, stream>>>(E, mu, rstd);
  k_gate_partial<<<dim3(BSc), 256, 0, stream>>>(E, mu, rstd, gamma, beta, We, partial);
  k_gate_reduce<<<dim3(Bc, Kc), 256, 0, stream>>>(partial, logits);
  k_softmax<<<1, 32, 0, stream>>>(logits, tau, gates);
  k_out<<<dim3((BSc * OUTc) / 256), 256, 0, stream>>>(E, gates, out);
}